// NodeFeat_77841987272986
// MI455X (gfx1250) — compile-verified
//
#include <hip/hip_runtime.h>
#include <cstdint>

#define D 64
#define WAVES 4

// ---------------- CSR build ----------------

__global__ void zero_counts_k(int* __restrict__ counts, int N) {
  int i = blockIdx.x * blockDim.x + threadIdx.x;
  if (i < N) counts[i] = 0;
}

__global__ void count_k(const int* __restrict__ row, int* __restrict__ counts, int E) {
  int e = blockIdx.x * blockDim.x + threadIdx.x;
  if (e < E) atomicAdd(&counts[row[e]], 1);
}

// Single-block (1024-thread) exclusive scan over counts -> rp[0..N], copy to nxt.
__global__ void scan_k(const int* __restrict__ counts, int* __restrict__ rp,
                       int* __restrict__ nxt, int N) {
  __shared__ int sums[1024];
  __shared__ int offs[1024];
  int t = threadIdx.x;
  int C = (N + 1023) >> 10;
  int b = t * C;
  int e2 = b + C;
  if (b > N) b = N;
  if (e2 > N) e2 = N;
  int s = 0;
  for (int i = b; i < e2; ++i) s += counts[i];
  sums[t] = s;
  __syncthreads();
  if (t == 0) {
    int r = 0;
    for (int k = 0; k < 1024; ++k) { offs[k] = r; r += sums[k]; }
    rp[N] = r;
  }
  __syncthreads();
  int r = offs[t];
  for (int i = b; i < e2; ++i) { rp[i] = r; nxt[i] = r; r += counts[i]; }
}

__global__ void scatter_k(const int* __restrict__ row, const int* __restrict__ col,
                          int* __restrict__ nxt, int* __restrict__ ec, int E) {
  int e = blockIdx.x * blockDim.x + threadIdx.x;
  if (e < E) {
    int r = row[e];
    int pos = atomicAdd(&nxt[r], 1);
    ec[pos] = col[e];
  }
}

// ---------------- Hop 1: y1 = (1/deg) * A^T x3, factored async gather ----------------
// x3[j, blk*64+d] = x[j,d] * {1, rsqrt(deg_j), sqrt(deg_j)}[blk], so per edge we only
// need x[j] (256B) + deg[j]. Two edges = 512B = 32 x 16B chunks = ONE async b128 per wave.

__device__ __forceinline__ void issue_xpair_async(const float* __restrict__ x,
                                                  int j0, int j1,
                                                  unsigned int ldsbase, int lane) {
  const char* r0 = (const char*)(x + (size_t)j0 * D);
  const char* r1 = (const char*)(x + (size_t)j1 * D);
  const char* ga = (lane < 16) ? (r0 + lane * 16) : (r1 + (lane - 16) * 16);
  unsigned long long gav = (unsigned long long)(uintptr_t)ga;
  unsigned int la = ldsbase + (unsigned int)lane * 16u;
  asm volatile("global_load_async_to_lds_b128 %0, %1, off"
               :: "v"(la), "v"(gav) : "memory");
}

__global__ __launch_bounds__(WAVES * 32) void spmm1_k(
    const float* __restrict__ x, const float* __restrict__ deg,
    const int* __restrict__ rp, const int* __restrict__ ec,
    float* __restrict__ y1, int N) {
  __shared__ __align__(16) float xbuf[WAVES][2][128];  // 2 x 512B per wave
  int lane = threadIdx.x & 31;
  int wid  = threadIdx.x >> 5;
  int i = blockIdx.x * WAVES + wid;
  if (i >= N) return;
  unsigned int lds0 = (unsigned int)(uintptr_t)&xbuf[wid][0][0];
  unsigned int lds1 = (unsigned int)(uintptr_t)&xbuf[wid][1][0];
  int beg = rp[i], end = rp[i + 1];
  int nedge = end - beg;
  int npairs = nedge >> 1;
  int c0 = 2 * lane;  // this lane owns features c0, c0+1
  float a0x = 0.f, a0y = 0.f, a1x = 0.f, a1y = 0.f, a2x = 0.f, a2y = 0.f;

  float d0 = 1.f, d1 = 1.f;
  if (npairs > 0) {
    int j0 = ec[beg], j1 = ec[beg + 1];
    issue_xpair_async(x, j0, j1, lds0, lane);
    d0 = deg[j0];
    d1 = deg[j1];
  }
  for (int p = 0; p < npairs; ++p) {
    float dn0 = 0.f, dn1 = 0.f;
    if (p + 1 < npairs) {
      int e = beg + 2 * (p + 1);
      int j0 = ec[e], j1 = ec[e + 1];
      issue_xpair_async(x, j0, j1, (p & 1) ? lds0 : lds1, lane);
      dn0 = deg[j0];
      dn1 = deg[j1];
      asm volatile("s_wait_asynccnt 0x1" ::: "memory");  // pair p landed (in-order)
    } else {
      asm volatile("s_wait_asynccnt 0x0" ::: "memory");
    }
    float rs0 = rsqrtf(d0), sq0 = sqrtf(d0);
    float rs1 = rsqrtf(d1), sq1 = sqrtf(d1);
    const float* cb = &xbuf[wid][p & 1][0];
    float2 v0 = *(const float2*)(cb + c0);        // edge 2p
    float2 v1 = *(const float2*)(cb + 64 + c0);   // edge 2p+1
    a0x += v0.x + v1.x;             a0y += v0.y + v1.y;
    a1x += v0.x * rs0 + v1.x * rs1; a1y += v0.y * rs0 + v1.y * rs1;
    a2x += v0.x * sq0 + v1.x * sq1; a2y += v0.y * sq0 + v1.y * sq1;
    d0 = dn0;
    d1 = dn1;
  }
  if (nedge & 1) {  // odd tail edge: direct gather
    int j = ec[end - 1];
    float2 xv = *(const float2*)(x + (size_t)j * D + c0);
    float dj = deg[j];
    float rs = rsqrtf(dj), sq = sqrtf(dj);
    a0x += xv.x;      a0y += xv.y;
    a1x += xv.x * rs; a1y += xv.y * rs;
    a2x += xv.x * sq; a2y += xv.y * sq;
  }
  float inv = 1.0f / deg[i];
  float* yr = y1 + (size_t)i * 192;
  *(float2*)(yr + c0)       = make_float2(a0x * inv, a0y * inv);
  *(float2*)(yr + 64 + c0)  = make_float2(a1x * inv, a1y * inv);
  *(float2*)(yr + 128 + c0) = make_float2(a2x * inv, a2y * inv);
}

// ---------------- Hop 2: async-to-LDS gather of full 768B y1 rows ----------------
// Two edges per step = 1536B = 96 x 16B chunks; lane l issues chunks l, l+32, l+64
// as global_load_async_to_lds_b128, double-buffered, pipelined via ASYNCcnt.

__device__ __forceinline__ void issue_ypair_async(const float* __restrict__ y1,
                                                  int j0, int j1,
                                                  unsigned int ldsbase, int lane) {
  const char* r0 = (const char*)(y1 + (size_t)j0 * 192);
  const char* r1 = (const char*)(y1 + (size_t)j1 * 192);
#pragma unroll
  for (int k = 0; k < 3; ++k) {
    int c = lane + 32 * k;                                  // chunk id in [0,96)
    const char* ga = (c < 48) ? (r0 + c * 16) : (r1 + (c - 48) * 16);
    unsigned long long gav = (unsigned long long)(uintptr_t)ga;
    unsigned int la = ldsbase + (unsigned int)c * 16u;
    asm volatile("global_load_async_to_lds_b128 %0, %1, off"
                 :: "v"(la), "v"(gav) : "memory");
  }
}

__global__ __launch_bounds__(WAVES * 32) void spmm2_k(
    const float* __restrict__ x, const float* __restrict__ deg,
    const int* __restrict__ rp, const int* __restrict__ ec,
    const float* __restrict__ y1, float* __restrict__ out, int N) {
  __shared__ __align__(16) float buf[WAVES][2][384];  // 2 x 1536B per wave
  int lane = threadIdx.x & 31;
  int wid  = threadIdx.x >> 5;
  int i = blockIdx.x * WAVES + wid;
  if (i >= N) return;
  unsigned int lds0 = (unsigned int)(uintptr_t)&buf[wid][0][0];
  unsigned int lds1 = (unsigned int)(uintptr_t)&buf[wid][1][0];
  int beg = rp[i], end = rp[i + 1];
  int nedge = end - beg;
  int npairs = nedge >> 1;
  int c0 = 2 * lane;
  float acc[3][2] = {{0.f, 0.f}, {0.f, 0.f}, {0.f, 0.f}};

  if (npairs > 0) issue_ypair_async(y1, ec[beg], ec[beg + 1], lds0, lane);
  for (int p = 0; p < npairs; ++p) {
    if (p + 1 < npairs) {
      int e = beg + 2 * (p + 1);
      issue_ypair_async(y1, ec[e], ec[e + 1], (p & 1) ? lds0 : lds1, lane);
      asm volatile("s_wait_asynccnt 0x3" ::: "memory");  // pair p landed (in-order)
    } else {
      asm volatile("s_wait_asynccnt 0x0" ::: "memory");
    }
    const float* cb = &buf[wid][p & 1][0];
#pragma unroll
    for (int k = 0; k < 3; ++k) {
      float2 v0 = *(const float2*)(cb + 64 * k + c0);        // edge 2p
      float2 v1 = *(const float2*)(cb + 192 + 64 * k + c0);  // edge 2p+1
      acc[k][0] += v0.x + v1.x;
      acc[k][1] += v0.y + v1.y;
    }
  }
  if (nedge & 1) {  // odd tail edge: direct gather
    int j = ec[end - 1];
#pragma unroll
    for (int k = 0; k < 3; ++k) {
      float2 v = *(const float2*)(y1 + (size_t)j * 192 + 64 * k + c0);
      acc[k][0] += v.x;
      acc[k][1] += v.y;
    }
  }

  // Finalize: recompute x3[i] on the fly, load y1[i], form y2, write 9 groups.
  float di  = deg[i];
  float inv = 1.0f / di;
  float rs  = rsqrtf(di);
  float sq  = sqrtf(di);
  float2 xv = *(const float2*)(x + (size_t)i * D + c0);
  float x3v[3][2] = {{xv.x, xv.y}, {xv.x * rs, xv.y * rs}, {xv.x * sq, xv.y * sq}};
  float y1v[3][2];
#pragma unroll
  for (int k = 0; k < 3; ++k) {
    float2 v = *(const float2*)(y1 + (size_t)i * 192 + 64 * k + c0);
    y1v[k][0] = v.x;
    y1v[k][1] = v.y;
  }
  float y2v[3][2];
#pragma unroll
  for (int k = 0; k < 3; ++k) {
    y2v[k][0] = acc[k][0] * inv - x3v[k][0];
    y2v[k][1] = acc[k][1] * inv - x3v[k][1];
  }
  float* o = out + (size_t)i * 576;
#pragma unroll
  for (int t = 0; t < 2; ++t) {
    float* q = o + (size_t)(c0 + t) * 9;  // out[i, d, 0..8], contiguous
    q[0] = x3v[0][t]; q[1] = x3v[1][t]; q[2] = x3v[2][t];
    q[3] = y1v[0][t]; q[4] = y1v[1][t]; q[5] = y1v[2][t];
    q[6] = y2v[0][t]; q[7] = y2v[1][t]; q[8] = y2v[2][t];
  }
}

// ---------------- host launcher ----------------

extern "C" void kernel_launch(void* const* d_in, const int* in_sizes, int n_in,
                              void* d_out, int out_size, void* d_ws, size_t ws_size,
                              hipStream_t stream) {
  const float* x   = (const float*)d_in[0];
  const float* deg = (const float*)d_in[1];
  const int*   row = (const int*)d_in[2];
  const int*   col = (const int*)d_in[3];
  float* out = (float*)d_out;
  int N = in_sizes[1];  // deg has N elements
  int E = in_sizes[2];  // row has E elements
  (void)n_in; (void)out_size; (void)ws_size;

  char* ws = (char*)d_ws;
  size_t off = 0;
  auto carve = [&](size_t bytes) -> void* {
    void* p = ws + off;
    off += (bytes + 255) & ~(size_t)255;
    return p;
  };
  float* y1   = (float*)carve((size_t)N * 192 * sizeof(float));  // 76.8 MB
  int*   cnts = (int*)carve((size_t)N * sizeof(int));
  int*   rp   = (int*)carve((size_t)(N + 1) * sizeof(int));
  int*   nxt  = (int*)carve((size_t)N * sizeof(int));
  int*   ec   = (int*)carve((size_t)E * sizeof(int));            // 6.4 MB

  const int tb = 256;
  zero_counts_k<<<(N + tb - 1) / tb, tb, 0, stream>>>(cnts, N);
  count_k<<<(E + tb - 1) / tb, tb, 0, stream>>>(row, cnts, E);
  scan_k<<<1, 1024, 0, stream>>>(cnts, rp, nxt, N);
  scatter_k<<<(E + tb - 1) / tb, tb, 0, stream>>>(row, col, nxt, ec, E);

  int blocks = (N + WAVES - 1) / WAVES;
  spmm1_k<<<blocks, WAVES * 32, 0, stream>>>(x, deg, rp, ec, y1, N);
  spmm2_k<<<blocks, WAVES * 32, 0, stream>>>(x, deg, rp, ec, y1, out, N);
}